// POP_22668837388598
// MI455X (gfx1250) — compile-verified
//
#include <hip/hip_runtime.h>
#include <hip/hip_bf16.h>

// Problem constants (from reference)
#define V_ITEMS   100001            // vocabulary (NUM_ITEMS + 1)
#define N_SORT    131072            // 2^17, next pow2 >= V_ITEMS
#define LOCAL_ELE 2048              // elements per LDS-fused bitonic block
#define LOCAL_THR 1024              // threads per LDS bitonic block (32 waves)

typedef unsigned long long u64;
typedef unsigned int       u32;

// Native clang vectors (accepted by nontemporal/async builtins; lower to b128)
typedef int   vi4 __attribute__((ext_vector_type(4)));
typedef float vf4 __attribute__((ext_vector_type(4)));

// Address-space-qualified pointer types for the CDNA5 async global<->LDS ops
typedef __attribute__((address_space(1))) vi4* gvi4p;   // global
typedef __attribute__((address_space(3))) vi4* lvi4p;   // LDS

// ---- CDNA5 async global<->LDS path (ASYNCcnt), guarded for toolchain support
#if __has_builtin(__builtin_amdgcn_global_load_async_to_lds_b128) && \
    __has_builtin(__builtin_amdgcn_global_store_async_from_lds_b128)
#define POP_HAS_ASYNC_LDS 1
#else
#define POP_HAS_ASYNC_LDS 0
#endif

__device__ __forceinline__ void pop_wait_async0() {
#if __has_builtin(__builtin_amdgcn_s_wait_asynccnt)
    __builtin_amdgcn_s_wait_asynccnt(0);
#else
    asm volatile("s_wait_asynccnt 0x0" ::: "memory");
#endif
}

// ---------------------------------------------------------------- zero counts
__global__ void pop_zero_counts(u32* __restrict__ counts, int v) {
    int i = blockIdx.x * blockDim.x + threadIdx.x;
    if (i < v) counts[i] = 0u;
}

// ---------------------------------------------------------------- histogram
// 3.28M int32 ids -> 100001 bins. NT b128 loads (read-once stream);
// global atomics resolve in L2 (400KB table is L2-resident).
__global__ void pop_hist(const int* __restrict__ in, long n4,
                         u32* __restrict__ counts) {
    long stride = (long)gridDim.x * blockDim.x;
    for (long t = (long)blockIdx.x * blockDim.x + threadIdx.x; t < n4; t += stride) {
        vi4 v = __builtin_nontemporal_load(&((const vi4*)in)[t]);
        atomicAdd(&counts[v.x], 1u);
        atomicAdd(&counts[v.y], 1u);
        atomicAdd(&counts[v.z], 1u);
        atomicAdd(&counts[v.w], 1u);
    }
}

// ---------------------------------------------------------------- build keys
// key = (count << 32) | (0xFFFFFFFF - item). Descending sort => descending
// count, ties broken by ascending item id (matches JAX stable argsort).
// Padding keys are 0 (< any real key since low word >= 0xFFFFFFFF-100000).
__global__ void pop_build_keys(const u32* __restrict__ counts,
                               u64* __restrict__ keys) {
    int i = blockIdx.x * blockDim.x + threadIdx.x;   // grid covers N_SORT
    u64 k = 0ull;
    if (i < V_ITEMS)
        k = ((u64)counts[i] << 32) | (u64)(0xFFFFFFFFu - (u32)i);
    keys[i] = k;
}

// ---------------------------------------------------------------- bitonic: global stage
// One (k, j) compare-exchange stage, descending order overall.
__global__ void pop_bitonic_global(u64* __restrict__ keys, u32 k, u32 j) {
    u32 i = blockIdx.x * blockDim.x + threadIdx.x;
    u32 p = i ^ j;
    if (p > i) {
        u64 a = keys[i];
        u64 b = keys[p];
        bool up = ((i & k) == 0u);               // "up" region keeps larger first
        bool sw = up ? (a < b) : (a > b);
        if (sw) { keys[i] = b; keys[p] = a; }
    }
}

// ---------------------------------------------------------------- bitonic: LDS-fused stages
// Each block owns a contiguous 2048-key (16KB) chunk in LDS and runs all
// stages with j <= 1024 for k in [k_first, k_last]. Chunk in/out transfers use
// CDNA5 async global<->LDS b128 ops (ASYNCcnt) when available: the data never
// touches VGPRs. Direction computed from the GLOBAL index, so
// alternating-direction chunks fall out automatically.
__global__ __launch_bounds__(LOCAL_THR)
void pop_bitonic_shared(u64* __restrict__ keys, u32 k_first, u32 k_last) {
    __shared__ __align__(16) u64 s[LOCAL_ELE];
    const u32 base = blockIdx.x * LOCAL_ELE;

#if POP_HAS_ASYNC_LDS
    // lane t moves 16B: 1024 lanes cover the whole 16KB chunk
    __builtin_amdgcn_global_load_async_to_lds_b128(
        (gvi4p)(vi4*)(keys + base + (u32)threadIdx.x * 2u),
        (lvi4p)(vi4*)(&s[(u32)threadIdx.x * 2u]),
        0, 0);
    pop_wait_async0();
#else
    s[threadIdx.x]             = keys[base + threadIdx.x];
    s[threadIdx.x + LOCAL_THR] = keys[base + threadIdx.x + LOCAL_THR];
#endif
    __syncthreads();

    for (u32 k = k_first; k <= k_last && k != 0u; k <<= 1) {
        u32 jstart = (k >> 1) < (u32)LOCAL_THR ? (k >> 1) : (u32)LOCAL_THR;
        for (u32 j = jstart; j >= 1u; j >>= 1) {
#pragma unroll
            for (int h = 0; h < 2; ++h) {
                u32 l = threadIdx.x + (u32)h * LOCAL_THR;
                u32 p = l ^ j;
                if (p > l) {
                    u64 a = s[l];
                    u64 b = s[p];
                    bool up = (((base + l) & k) == 0u);
                    bool sw = up ? (a < b) : (a > b);
                    if (sw) { s[l] = b; s[p] = a; }
                }
            }
            __syncthreads();
        }
    }

#if POP_HAS_ASYNC_LDS
    __builtin_amdgcn_global_store_async_from_lds_b128(
        (gvi4p)(vi4*)(keys + base + (u32)threadIdx.x * 2u),
        (lvi4p)(vi4*)(&s[(u32)threadIdx.x * 2u]),
        0, 0);
    // S_ENDPGM performs an implicit wait-idle; next kernel sees completed stores.
#else
    keys[base + threadIdx.x]             = s[threadIdx.x];
    keys[base + threadIdx.x + LOCAL_THR] = s[threadIdx.x + LOCAL_THR];
#endif
}

// ---------------------------------------------------------------- scatter 1/rank
__global__ void pop_scatter_ranks(const u64* __restrict__ keys,
                                  float* __restrict__ inv_rank) {
    int p = blockIdx.x * blockDim.x + threadIdx.x;   // grid covers N_SORT
    u64 k = keys[p];
    if (k == 0ull) return;                           // padding
    u32 item = 0xFFFFFFFFu - (u32)k;
    if (item < (u32)V_ITEMS)
        inv_rank[item] = 1.0f / (float)(p + 1);      // 1-based rank
}

// ---------------------------------------------------------------- gather
// 16.38M random reads of a 400KB table (kept regular-temporal -> L2 resident).
// Index stream: NT b128 loads + global_prefetch_b8 one grid-stride ahead.
// Output: NT b128 stores (write-once, don't pollute caches).
__global__ void pop_gather(const int* __restrict__ idx,
                           const float* __restrict__ inv_rank,
                           float* __restrict__ out, long n4) {
    long stride = (long)gridDim.x * blockDim.x;
    for (long t = (long)blockIdx.x * blockDim.x + threadIdx.x; t < n4; t += stride) {
        if (t + stride < n4)
            __builtin_prefetch(&((const vi4*)idx)[t + stride], 0, 0);
        vi4 v = __builtin_nontemporal_load(&((const vi4*)idx)[t]);
        vf4 r;
        r.x = inv_rank[v.x];
        r.y = inv_rank[v.y];
        r.z = inv_rank[v.z];
        r.w = inv_rank[v.w];
        __builtin_nontemporal_store(r, &((vf4*)out)[t]);
    }
}

// ---------------------------------------------------------------- launcher
extern "C" void kernel_launch(void* const* d_in, const int* in_sizes, int n_in,
                              void* d_out, int out_size, void* d_ws, size_t ws_size,
                              hipStream_t stream) {
    const int*  input_seqs = (const int*)d_in[0];     // 16384*200  int32
    const int*  poss_seqs  = (const int*)d_in[1];     // 16384*1000 int32
    float*      out        = (float*)d_out;           // 16384*1000 f32
    (void)n_in; (void)ws_size;

    const long n_hist = (long)in_sizes[0];            // 3,276,800 (mult of 4)
    const long n_out  = (long)out_size;               // 16,384,000 (mult of 4)

    // Workspace layout (all within ~1.85 MB):
    //   [0)              counts   : V_ITEMS u32
    //   [400384)         keys     : N_SORT  u64   (16B aligned)
    //   [400384+1MiB)    inv_rank : V_ITEMS f32
    char* ws = (char*)d_ws;
    u32*   counts   = (u32*)(ws);
    u64*   keys     = (u64*)(ws + 400384);
    float* inv_rank = (float*)(ws + 400384 + (size_t)N_SORT * 8);

    // 1) clear counts (ws is poisoned by the harness)
    pop_zero_counts<<<(V_ITEMS + 255) / 256, 256, 0, stream>>>(counts, V_ITEMS);

    // 2) popularity histogram
    pop_hist<<<2048, 256, 0, stream>>>(input_seqs, n_hist / 4, counts);

    // 3) composite sort keys
    pop_build_keys<<<N_SORT / 256, 256, 0, stream>>>(counts, keys);

    // 4) descending bitonic sort of N_SORT 64-bit keys.
    //    k <= 2048 fully inside LDS; larger k: global stages down to j=2048,
    //    then one LDS-fused pass for j <= 1024.
    pop_bitonic_shared<<<N_SORT / LOCAL_ELE, LOCAL_THR, 0, stream>>>(keys, 2u, 2048u);
    for (u32 k = 4096u; k <= (u32)N_SORT; k <<= 1) {
        for (u32 j = k >> 1; j >= 2048u; j >>= 1)
            pop_bitonic_global<<<N_SORT / 256, 256, 0, stream>>>(keys, k, j);
        pop_bitonic_shared<<<N_SORT / LOCAL_ELE, LOCAL_THR, 0, stream>>>(keys, k, k);
    }

    // 5) rank -> reciprocal-rank table
    pop_scatter_ranks<<<N_SORT / 256, 256, 0, stream>>>(keys, inv_rank);

    // 6) bandwidth-bound gather into the output
    pop_gather<<<4096, 256, 0, stream>>>(poss_seqs, inv_rank, out, n_out / 4);
}